// GATLayer_43138651521644
// MI455X (gfx1250) — compile-verified
//
#include <hip/hip_runtime.h>
#include <hip/hip_bf16.h>
#include <math.h>

typedef __attribute__((ext_vector_type(2))) float v2f;
typedef __attribute__((ext_vector_type(8))) float v8f;

#define IN_F   256
#define OUT_F  128
#define NHEAD  4
#define COLS   (NHEAD * OUT_F)   // 512
#define NEG_SLOPE 0.2f
#define MAXBLK_E  704            // blocks for edge-max partials
#define KSTEPS    (IN_F / 4)     // 64
#define NCT       32             // column tiles = H*OUT_F/16

// ---------------------------------------------------------------------------
// K0: repack W [H,256,128] into WMMA-B fragment order so each B fragment is a
// single coalesced global_load_b64.
//   Wt (as v2f): [ct][kstep][lane] , ct = head*8 + otile, kstep = k/4
//   lane l -> col = l&15, kb = 2*(l>>4); v2f = { W[k+kb][o0+col], W[k+kb+1][o0+col] }
// ---------------------------------------------------------------------------
__global__ __launch_bounds__(256) void gat_repack_w(
    const float* __restrict__ W, float* __restrict__ Wt) {
  const int idx = blockIdx.x * blockDim.x + threadIdx.x;  // over 32*64*32 = 65536
  if (idx >= NCT * KSTEPS * 32) return;
  const int lane  = idx & 31;
  const int kstep = (idx >> 5) & (KSTEPS - 1);
  const int ct    = idx >> 11;                 // 0..31
  const int head  = ct >> 3;
  const int o     = (ct & 7) * 16 + (lane & 15);
  const int k     = kstep * 4 + 2 * (lane >> 4);
  const float* wb = W + (size_t)head * (IN_F * OUT_F) + o;
  Wt[(size_t)idx * 2 + 0] = wb[(size_t)k * OUT_F];
  Wt[(size_t)idx * 2 + 1] = wb[(size_t)(k + 1) * OUT_F];
}

// ---------------------------------------------------------------------------
// K1: x [N,256] @ W[h] [256,128] -> out [N, 512] (x_t in final layout)
// One 32-node tile per block; 8 waves; each wave owns 4 of the 32 column
// tiles and 2 M-subtiles -> 8 accumulators, A fragments reused 4x, B 2x.
// V_WMMA_F32_16X16X4_F32 fragment layouts (wave32):
//   A 16x4 : lane l -> M = l&15, K = 2*(l>>4) + {0,1}
//   B 4x16 : lane l -> Ncol = l&15, K = 2*(l>>4) + {0,1}
//   C 16x16: vgpr g -> M = g + 8*(l>>4), Ncol = l&15
// ---------------------------------------------------------------------------
__global__ __launch_bounds__(256) void gat_gemm_wmma(
    const float* __restrict__ x, const float* __restrict__ Wt,
    float* __restrict__ out, int n_nodes) {
  const int n0   = blockIdx.x * 32;
  const int lane = threadIdx.x & 31;
  const int wave = threadIdx.x >> 5;        // 0..7
  const int m    = lane & 15;
  const int kb   = (lane >> 4) * 2;         // 0 or 2

  // clamp rows (no branch): garbage rows only affect guarded output rows
  const int r0 = min(n0 + m,      n_nodes - 1);
  const int r1 = min(n0 + 16 + m, n_nodes - 1);
  const float* xrow0 = x + (size_t)r0 * IN_F + kb;
  const float* xrow1 = x + (size_t)r1 * IN_F + kb;

  // B fragment streams for this wave's 4 column tiles
  const v2f* bt[4];
#pragma unroll
  for (int j = 0; j < 4; ++j)
    bt[j] = (const v2f*)Wt + (size_t)(wave + 8 * j) * (KSTEPS * 32) + lane;

  v8f acc[2][4] = {};

  for (int ks = 0; ks < KSTEPS; ++ks) {
    const v2f a0 = *(const v2f*)(xrow0 + ks * 4);
    const v2f a1 = *(const v2f*)(xrow1 + ks * 4);
#pragma unroll
    for (int j = 0; j < 4; ++j) {
      const v2f b = bt[j][(size_t)ks * 32];
      acc[0][j] = __builtin_amdgcn_wmma_f32_16x16x4_f32(
          false, a0, false, b, (short)0, acc[0][j], false, false);
      acc[1][j] = __builtin_amdgcn_wmma_f32_16x16x4_f32(
          false, a1, false, b, (short)0, acc[1][j], false, false);
    }
  }

  const int col = lane & 15;
#pragma unroll
  for (int t = 0; t < 2; ++t) {
#pragma unroll
    for (int j = 0; j < 4; ++j) {
      const int ct   = wave + 8 * j;
      const int head = ct >> 3;
      const int o0   = (ct & 7) * 16;
#pragma unroll
      for (int g = 0; g < 8; ++g) {
        const int r = n0 + t * 16 + g + (lane >> 4) * 8;
        if (r < n_nodes)
          out[(size_t)r * COLS + head * OUT_F + o0 + col] = acc[t][j][g];
      }
    }
  }
}

// ---------------------------------------------------------------------------
// K2: s_src[h,n] = xt[h,n,:].a[h,:128], s_dst[h,n] = xt[h,n,:].a[h,128:256]
// One wave per (h,n); lane loads one float4 (whole 512B row covered).
// ---------------------------------------------------------------------------
__global__ __launch_bounds__(256) void gat_scores(
    const float* __restrict__ xt /* [N,512] */, const float* __restrict__ a,
    float* __restrict__ s_src, float* __restrict__ s_dst, int n_nodes) {
  const int wid  = (int)((blockIdx.x * blockDim.x + threadIdx.x) >> 5);
  const int lane = threadIdx.x & 31;
  if (wid >= NHEAD * n_nodes) return;
  const int h = wid / n_nodes;
  const int n = wid - h * n_nodes;
  const float4 v  = *(const float4*)(xt + (size_t)n * COLS + h * OUT_F + lane * 4);
  const float4 va = *(const float4*)(a + (size_t)h * (2 * OUT_F) + lane * 4);
  const float4 vb = *(const float4*)(a + (size_t)h * (2 * OUT_F) + OUT_F + lane * 4);
  float ss = v.x * va.x + v.y * va.y + v.z * va.z + v.w * va.w;
  float sd = v.x * vb.x + v.y * vb.y + v.z * vb.z + v.w * vb.w;
#pragma unroll
  for (int off = 16; off > 0; off >>= 1) {
    ss += __shfl_down(ss, off, 32);
    sd += __shfl_down(sd, off, 32);
  }
  if (lane == 0) {
    s_src[(size_t)h * n_nodes + n] = ss;
    s_dst[(size_t)h * n_nodes + n] = sd;
  }
}

__device__ __forceinline__ float leaky(float v) {
  return v > 0.0f ? v : NEG_SLOPE * v;
}

// ---------------------------------------------------------------------------
// K3: per-block per-head max of leaky_relu(s_src[h,src]+s_dst[h,dst])
// ---------------------------------------------------------------------------
__global__ __launch_bounds__(256) void gat_edge_max(
    const int* __restrict__ src, const int* __restrict__ dst,
    const float* __restrict__ s_src, const float* __restrict__ s_dst,
    float* __restrict__ partial /* [NHEAD*gridDim.x] */, int n_nodes, int n_edges) {
  float mx[NHEAD];
#pragma unroll
  for (int h = 0; h < NHEAD; ++h) mx[h] = -INFINITY;
  for (int e = blockIdx.x * blockDim.x + threadIdx.x; e < n_edges;
       e += gridDim.x * blockDim.x) {
    const int s = src[e], d = dst[e];
#pragma unroll
    for (int h = 0; h < NHEAD; ++h) {
      const float v = leaky(s_src[(size_t)h * n_nodes + s] +
                            s_dst[(size_t)h * n_nodes + d]);
      mx[h] = fmaxf(mx[h], v);
    }
  }
  __shared__ float smax[NHEAD][8];
  const int lane = threadIdx.x & 31, wave = threadIdx.x >> 5;
#pragma unroll
  for (int h = 0; h < NHEAD; ++h) {
#pragma unroll
    for (int off = 16; off > 0; off >>= 1)
      mx[h] = fmaxf(mx[h], __shfl_down(mx[h], off, 32));
    if (lane == 0) smax[h][wave] = mx[h];
  }
  __syncthreads();
  if (threadIdx.x < NHEAD) {
    float m = -INFINITY;
#pragma unroll
    for (int w = 0; w < 8; ++w) m = fmaxf(m, smax[threadIdx.x][w]);
    partial[(size_t)threadIdx.x * gridDim.x + blockIdx.x] = m;
  }
}

__global__ __launch_bounds__(256) void gat_reduce_max(
    const float* __restrict__ partial, float* __restrict__ headmax, int nparts) {
  const int h = blockIdx.x;
  float m = -INFINITY;
  for (int i = threadIdx.x; i < nparts; i += blockDim.x)
    m = fmaxf(m, partial[(size_t)h * nparts + i]);
  __shared__ float smax[8];
  const int lane = threadIdx.x & 31, wave = threadIdx.x >> 5;
#pragma unroll
  for (int off = 16; off > 0; off >>= 1) m = fmaxf(m, __shfl_down(m, off, 32));
  if (lane == 0) smax[wave] = m;
  __syncthreads();
  if (threadIdx.x == 0) {
#pragma unroll
    for (int w = 1; w < 8; ++w) m = fmaxf(m, smax[w]);
    headmax[h] = m;
  }
}

// ---------------------------------------------------------------------------
// K4: denom[h,dst] += exp(e - headmax[h])      (f32 global atomics)
// ---------------------------------------------------------------------------
__global__ __launch_bounds__(256) void gat_denom(
    const int* __restrict__ src, const int* __restrict__ dst,
    const float* __restrict__ s_src, const float* __restrict__ s_dst,
    const float* __restrict__ headmax, float* __restrict__ denom,
    int n_nodes, int n_edges) {
  const int e = blockIdx.x * blockDim.x + threadIdx.x;
  if (e >= n_edges) return;
  const int s = src[e], d = dst[e];
#pragma unroll
  for (int h = 0; h < NHEAD; ++h) {
    const float v = leaky(s_src[(size_t)h * n_nodes + s] +
                          s_dst[(size_t)h * n_nodes + d]);
    atomicAdd(&denom[(size_t)h * n_nodes + d], expf(v - headmax[h]));
  }
}

// ---------------------------------------------------------------------------
// K5: self-loop edges only: diag[h,src] += w / (denom[h,dst] + 1e-16)
// ---------------------------------------------------------------------------
__global__ __launch_bounds__(256) void gat_diag(
    const int* __restrict__ src, const int* __restrict__ dst,
    const float* __restrict__ s_src, const float* __restrict__ s_dst,
    const float* __restrict__ headmax, const float* __restrict__ denom,
    float* __restrict__ diag, int n_nodes, int n_edges) {
  const int e = blockIdx.x * blockDim.x + threadIdx.x;
  if (e >= n_edges) return;
  const int s = src[e], d = dst[e];
  if (s != d) return;
#pragma unroll
  for (int h = 0; h < NHEAD; ++h) {
    const float v = leaky(s_src[(size_t)h * n_nodes + s] +
                          s_dst[(size_t)h * n_nodes + d]);
    const float w = expf(v - headmax[h]);
    const float alpha = w / (denom[(size_t)h * n_nodes + d] + 1e-16f);
    atomicAdd(&diag[(size_t)h * n_nodes + s], alpha);
  }
}

// ---------------------------------------------------------------------------
// K6: out[n, h*128+o] *= diag[h,n]   (in-place scale of x_t, float4 wide)
// ---------------------------------------------------------------------------
__global__ __launch_bounds__(256) void gat_scale(
    float* __restrict__ out, const float* __restrict__ diag, int n_nodes) {
  const size_t idx = (size_t)blockIdx.x * blockDim.x + threadIdx.x;  // float4 index
  const size_t total = (size_t)n_nodes * (COLS / 4);
  if (idx >= total) return;
  const int n = (int)(idx / (COLS / 4));
  const int c4 = (int)(idx - (size_t)n * (COLS / 4));
  const int h = c4 >> 5;  // (c4*4)/128
  const float s = diag[(size_t)h * n_nodes + n];
  float4* p = (float4*)out + idx;
  float4 v = *p;
  v.x *= s; v.y *= s; v.z *= s; v.w *= s;
  *p = v;
}

// ---------------------------------------------------------------------------
extern "C" void kernel_launch(void* const* d_in, const int* in_sizes, int n_in,
                              void* d_out, int out_size, void* d_ws, size_t ws_size,
                              hipStream_t stream) {
  const float* x  = (const float*)d_in[0];   // [N, 256]
  const float* W  = (const float*)d_in[1];   // [4, 256, 128]
  const float* a  = (const float*)d_in[2];   // [4, 256, 1]
  const int* eidx = (const int*)d_in[3];     // [2, E]
  float* out = (float*)d_out;                // [N, 512]

  const int n_nodes = in_sizes[0] / IN_F;
  const int n_edges = in_sizes[3] / 2;
  const int* src = eidx;
  const int* dst = eidx + n_edges;

  float* ws      = (float*)d_ws;
  float* s_src   = ws;                              // H*N
  float* s_dst   = s_src + (size_t)NHEAD * n_nodes; // H*N
  float* denom   = s_dst + (size_t)NHEAD * n_nodes; // H*N
  float* diag    = denom + (size_t)NHEAD * n_nodes; // H*N
  float* headmax = diag  + (size_t)NHEAD * n_nodes; // H (pad 16)
  float* partial = headmax + 16;                    // H*MAXBLK_E
  float* Wt      = partial + (size_t)NHEAD * MAXBLK_E; // H*IN_F*OUT_F repacked W

  // zero denom + diag (contiguous)
  hipMemsetAsync(denom, 0, (size_t)2 * NHEAD * n_nodes * sizeof(float), stream);

  // K0: repack W into B-fragment order (one b64 per fragment afterwards)
  gat_repack_w<<<(NCT * KSTEPS * 32 + 255) / 256, 256, 0, stream>>>(W, Wt);

  // K1: WMMA GEMM, x_t written directly into d_out layout
  gat_gemm_wmma<<<(n_nodes + 31) / 32, 256, 0, stream>>>(x, Wt, out, n_nodes);

  // K2: per-(h,n) scores
  {
    const int nwaves = NHEAD * n_nodes;
    gat_scores<<<(nwaves + 7) / 8, 256, 0, stream>>>(out, a, s_src, s_dst, n_nodes);
  }

  // K3: per-head max over edges
  gat_edge_max<<<MAXBLK_E, 256, 0, stream>>>(src, dst, s_src, s_dst, partial,
                                             n_nodes, n_edges);
  gat_reduce_max<<<NHEAD, 256, 0, stream>>>(partial, headmax, MAXBLK_E);

  // K4: softmax denominators via f32 global atomics
  gat_denom<<<(n_edges + 255) / 256, 256, 0, stream>>>(
      src, dst, s_src, s_dst, headmax, denom, n_nodes, n_edges);

  // K5: self-loop diagonal attention
  gat_diag<<<(n_edges + 255) / 256, 256, 0, stream>>>(
      src, dst, s_src, s_dst, headmax, denom, diag, n_nodes, n_edges);

  // K6: in-place scale (float4 wide)
  {
    const size_t total = (size_t)n_nodes * (COLS / 4);
    gat_scale<<<(int)((total + 255) / 256), 256, 0, stream>>>(out, diag, n_nodes);
  }
}